// CausalSelfAttention_48120813584604
// MI455X (gfx1250) — compile-verified
//
#include <hip/hip_runtime.h>
#include <math.h>
#include <stdint.h>

// ---------------------------------------------------------------------------
// Causal self-attention forward for MI455X (gfx1250), fp32 WMMA path.
//   B=4, T=2048, C=1024, H=16, HD=64
// Outputs (flat, in order): y [B,T,C], k [B,H,T,HD], v [B,H,T,HD]
//
// GEMM kernels stage the shared B tile through LDS with CDNA5 async
// global->LDS loads (ASYNCcnt), double-buffered across K-chunks.
// ---------------------------------------------------------------------------

typedef __attribute__((ext_vector_type(2))) float v2f;
typedef __attribute__((ext_vector_type(8))) float v8f;

#define BATCH 4
#define SEQ   2048
#define CDIM  1024
#define NCOL  3072
#define NHEAD 16
#define HDIM  64
#define MROWS (BATCH * SEQ)

__device__ __forceinline__ v8f wmma4(v2f a, v2f b, v8f c) {
  // D = A(16x4, f32) * B(4x16, f32) + C(16x16, f32)
  return __builtin_amdgcn_wmma_f32_16x16x4_f32(false, a, false, b, (short)0, c,
                                               false, false);
}

// Async global->LDS copy of one 16-byte packet per thread (ASYNCcnt-tracked).
// ldsAddr = low 32 bits of the flat shared pointer == DS-space byte address.
__device__ __forceinline__ void async_copy_b128(float* ldsDst,
                                                const float* gSrc) {
  unsigned lds = (unsigned)(uintptr_t)ldsDst;
  asm volatile("global_load_async_to_lds_b128 %0, %1, off"
               :: "v"(lds), "v"(gSrc)
               : "memory");
}

__device__ __forceinline__ void wait_async0() {
  asm volatile("s_wait_asynccnt 0" ::: "memory");
}

// Stage a 16(K) x 64(N) B chunk into LDS: 256 threads x one b128 async load.
__device__ __forceinline__ void stage_b_chunk(const float* __restrict__ w,
                                              int ldw, int k0, int nBase,
                                              float* dstLds) {
  const int t    = threadIdx.x;
  const int row  = t >> 4;         // 0..15  (K within chunk)
  const int colq = (t & 15) * 4;   // 0..60  (N, quad of floats)
  const float* g = w + (size_t)(k0 + row) * ldw + nBase + colq;
  async_copy_b128(dstLds + row * 64 + colq, g);
}

// ---------------------------------------------------------------------------
// Kernel 1: qkv = x @ w_attn + b_attn ; q,k += pos_emb ; head-split scatter.
// Block = 256 threads = 8 waves; wave computes a 16(M) x 64(N) strip; the
// 64-col B tile is shared by all 8 waves via async-staged LDS double buffer.
// Grid = (NCOL/64, MROWS/128).
// ---------------------------------------------------------------------------
__global__ __launch_bounds__(256) void qkv_kernel(
    const float* __restrict__ x, const float* __restrict__ pos,
    const float* __restrict__ w, const float* __restrict__ bias,
    float* __restrict__ qh, float* __restrict__ kh, float* __restrict__ vh) {
  __shared__ float bsh[2][16 * 64];  // double-buffered 16x64 B chunk (8 KB)

  const int lane = threadIdx.x & 31;
  const int wv   = threadIdx.x >> 5;
  const int half = lane >> 4;   // K-half select for A/B fragments
  const int l16  = lane & 15;   // M row (A) / N col (B,C)

  const int mBase = blockIdx.y * 128 + wv * 16;
  const int nBase = blockIdx.x * 64;

  // A fragment source: lanes 0-15 hold rows M=0..15 at K=k,k+1;
  // lanes 16-31 hold the same rows at K=k+2,k+3.
  const float* aRow = x + (size_t)(mBase + l16) * CDIM + half * 2;

  v8f acc[4] = {};

  stage_b_chunk(w, NCOL, 0, nBase, bsh[0]);
  wait_async0();
  __syncthreads();

  for (int kc = 0; kc < CDIM / 16; ++kc) {
    const int cur = kc & 1;
    if (kc + 1 < CDIM / 16)
      stage_b_chunk(w, NCOL, (kc + 1) * 16, nBase, bsh[cur ^ 1]);

    const float* bs = bsh[cur];
    const int k0 = kc * 16;
#pragma unroll
    for (int kk = 0; kk < 4; ++kk) {
      v2f a = *(const v2f*)(aRow + k0 + kk * 4);
      const int ksl = kk * 4 + half * 2;
#pragma unroll
      for (int j = 0; j < 4; ++j) {
        v2f b;
        b.x = bs[ksl * 64 + j * 16 + l16];        // conflict-free ds reads
        b.y = bs[(ksl + 1) * 64 + j * 16 + l16];
        acc[j] = wmma4(a, b, acc[j]);             // A reused across 4 N tiles
      }
    }
    wait_async0();     // own wave's prefetch landed
    __syncthreads();   // everyone done reading the buffer being refilled next
  }

  // Epilogue: C/D layout -> M = mBase + r + half*8, N = nBase + j*16 + l16.
#pragma unroll
  for (int j = 0; j < 4; ++j) {
#pragma unroll
    for (int r = 0; r < 8; ++r) {
      const int row = mBase + r + half * 8;
      const int col = nBase + j * 16 + l16;
      const int bb    = row >> 11;          // row / SEQ
      const int t     = row & (SEQ - 1);
      const int which = col >> 10;          // 0=q,1=k,2=v
      const int c     = col & (CDIM - 1);
      const int h     = c >> 6;
      const int d     = c & (HDIM - 1);
      float val = acc[j][r] + bias[col];
      if (which < 2) val += pos[t * CDIM + c];  // pos_emb on q and k
      const size_t dst = (((size_t)bb * NHEAD + h) * SEQ + t) * HDIM + d;
      if (which == 0)      qh[dst] = val;
      else if (which == 1) kh[dst] = val;
      else                 vh[dst] = val;
    }
  }
}

// ---------------------------------------------------------------------------
// Kernel 2: streaming (flash) causal attention, one wave per 16-row q tile.
// Grid = B*H*(T/16)/8 blocks of 256 threads.
// ---------------------------------------------------------------------------
__global__ __launch_bounds__(256) void attn_kernel(
    const float* __restrict__ qh, const float* __restrict__ kh,
    const float* __restrict__ vh, float* __restrict__ yh) {
  // Per-wave 16x16 P staging tile, stride 18 floats: conflict-free and
  // 8-byte aligned for b64 fragment reloads.
  __shared__ float pbuf[8 * 16 * 18];

  const int lane = threadIdx.x & 31;
  const int wv   = threadIdx.x >> 5;
  const int half = lane >> 4;
  const int l16  = lane & 15;

  const int wid   = blockIdx.x * 8 + wv;
  const int bh    = wid >> 7;    // 128 q tiles per (b,h)
  const int qi    = wid & 127;
  const int qBase = qi * 16;

  const float* Q = qh + (size_t)bh * SEQ * HDIM;
  const float* K = kh + (size_t)bh * SEQ * HDIM;
  const float* V = vh + (size_t)bh * SEQ * HDIM;
  float* pw = pbuf + wv * (16 * 18);

  // Cache all Q fragments for this tile: 16 k-steps of K=4.
  v2f qa[16];
  {
    const float* qRow = Q + (size_t)(qBase + l16) * HDIM + half * 2;
#pragma unroll
    for (int kk = 0; kk < 16; ++kk) qa[kk] = *(const v2f*)(qRow + kk * 4);
  }

  v8f o[4] = {};
  float mrow[8], lrow[8];
#pragma unroll
  for (int r = 0; r < 8; ++r) { mrow[r] = -INFINITY; lrow[r] = 0.0f; }

  for (int kt = 0; kt <= qi; ++kt) {
    const int kBase = kt * 16;

    // S = Q @ K^T  (B fragment: lane col = key row, contiguous HD loads)
    v8f s = {};
    const float* kRow = K + (size_t)(kBase + l16) * HDIM + half * 2;
#pragma unroll
    for (int kk = 0; kk < 16; ++kk) {
      v2f b = *(const v2f*)(kRow + kk * 4);
      s = wmma4(qa[kk], b, s);
    }

    // Scale, causal mask, online softmax (row reductions within 16-lane half)
    float alpha[8];
#pragma unroll
    for (int r = 0; r < 8; ++r) {
      float sv = s[r] * 0.125f;  // 1/sqrt(64)
      if (kt == qi && l16 > (r + half * 8)) sv = -INFINITY;
      float mx = sv;
#pragma unroll
      for (int off = 8; off >= 1; off >>= 1)
        mx = fmaxf(mx, __shfl_xor(mx, off, 32));
      const float mnew = fmaxf(mrow[r], mx);
      const float a    = __expf(mrow[r] - mnew);
      const float p    = __expf(sv - mnew);
      float ps = p;
#pragma unroll
      for (int off = 8; off >= 1; off >>= 1)
        ps += __shfl_xor(ps, off, 32);
      lrow[r]  = lrow[r] * a + ps;
      mrow[r]  = mnew;
      alpha[r] = a;
      // stash P in LDS (C layout: row = r + half*8, col = l16)
      pw[(r + half * 8) * 18 + l16] = p;
    }

    // Rescale running output
#pragma unroll
    for (int j = 0; j < 4; ++j)
#pragma unroll
      for (int r = 0; r < 8; ++r) o[j][r] *= alpha[r];

    // O += P @ V  (A fragment re-read from LDS in A layout)
    const float* vB = V + (size_t)kBase * HDIM;
#pragma unroll
    for (int kk = 0; kk < 4; ++kk) {
      const int ks = kk * 4 + half * 2;
      v2f a = *(const v2f*)(pw + l16 * 18 + ks);
#pragma unroll
      for (int j = 0; j < 4; ++j) {
        v2f b;
        b.x = vB[(size_t)ks * HDIM + j * 16 + l16];
        b.y = vB[(size_t)(ks + 1) * HDIM + j * 16 + l16];
        o[j] = wmma4(a, b, o[j]);
      }
    }
  }

  float inv[8];
#pragma unroll
  for (int r = 0; r < 8; ++r) inv[r] = 1.0f / lrow[r];
#pragma unroll
  for (int j = 0; j < 4; ++j)
#pragma unroll
    for (int r = 0; r < 8; ++r) {
      const int row = qBase + r + half * 8;
      const int col = j * 16 + l16;
      yh[((size_t)bh * SEQ + row) * HDIM + col] = o[j][r] * inv[r];
    }
}

// ---------------------------------------------------------------------------
// Kernel 3: y = merge_heads(y_heads) @ w_proj + b_proj.
// Same async-staged GEMM skeleton; A gathered from head-split layout.
// Grid = (CDIM/64, MROWS/128).
// ---------------------------------------------------------------------------
__global__ __launch_bounds__(256) void proj_kernel(
    const float* __restrict__ yhead, const float* __restrict__ w,
    const float* __restrict__ bias, float* __restrict__ out) {
  __shared__ float bsh[2][16 * 64];

  const int lane = threadIdx.x & 31;
  const int wv   = threadIdx.x >> 5;
  const int half = lane >> 4;
  const int l16  = lane & 15;

  const int mBase = blockIdx.y * 128 + wv * 16;
  const int nBase = blockIdx.x * 64;

  const int row = mBase + l16;
  const int bb  = row >> 11;
  const int t   = row & (SEQ - 1);
  // yhead[((bb*H + h)*T + t)*HD + d], with k = h*64 + d
  const float* aBase =
      yhead + (size_t)bb * NHEAD * SEQ * HDIM + (size_t)t * HDIM;

  v8f acc[4] = {};

  stage_b_chunk(w, CDIM, 0, nBase, bsh[0]);
  wait_async0();
  __syncthreads();

  for (int kc = 0; kc < CDIM / 16; ++kc) {
    const int cur = kc & 1;
    if (kc + 1 < CDIM / 16)
      stage_b_chunk(w, CDIM, (kc + 1) * 16, nBase, bsh[cur ^ 1]);

    const float* bs = bsh[cur];
    const int k0 = kc * 16;
#pragma unroll
    for (int kk = 0; kk < 4; ++kk) {
      const int ks = k0 + kk * 4 + half * 2;  // even -> ks,ks+1 in same head
      v2f a =
          *(const v2f*)(aBase + (size_t)(ks >> 6) * SEQ * HDIM + (ks & 63));
      const int ksl = kk * 4 + half * 2;
#pragma unroll
      for (int j = 0; j < 4; ++j) {
        v2f b;
        b.x = bs[ksl * 64 + j * 16 + l16];
        b.y = bs[(ksl + 1) * 64 + j * 16 + l16];
        acc[j] = wmma4(a, b, acc[j]);
      }
    }
    wait_async0();
    __syncthreads();
  }

#pragma unroll
  for (int j = 0; j < 4; ++j)
#pragma unroll
    for (int r = 0; r < 8; ++r) {
      const int orow = mBase + r + half * 8;
      const int col  = nBase + j * 16 + l16;
      out[(size_t)orow * CDIM + col] = acc[j][r] + bias[col];
    }
}

// ---------------------------------------------------------------------------
extern "C" void kernel_launch(void* const* d_in, const int* in_sizes, int n_in,
                              void* d_out, int out_size, void* d_ws,
                              size_t ws_size, hipStream_t stream) {
  const float* x   = (const float*)d_in[0];
  const float* pos = (const float*)d_in[1];
  const float* wat = (const float*)d_in[2];
  const float* bat = (const float*)d_in[3];
  const float* wpr = (const float*)d_in[4];
  const float* bpr = (const float*)d_in[5];

  float* out = (float*)d_out;
  float* y   = out;                                        // [B,T,C]
  float* kh  = out + (size_t)BATCH * SEQ * CDIM;           // [B,H,T,HD]
  float* vh  = kh + (size_t)BATCH * NHEAD * SEQ * HDIM;    // [B,H,T,HD]

  float* qh  = (float*)d_ws;                               // [B,H,T,HD] 33.5MB
  float* yhd = qh + (size_t)BATCH * NHEAD * SEQ * HDIM;    // [B,H,T,HD] 33.5MB

  dim3 blk(256);
  qkv_kernel<<<dim3(NCOL / 64, MROWS / 128), blk, 0, stream>>>(
      x, pos, wat, bat, qh, kh, vh);
  attn_kernel<<<dim3((BATCH * NHEAD * (SEQ / 16)) / 8), blk, 0, stream>>>(
      qh, kh, vh, yhd);
  proj_kernel<<<dim3(CDIM / 64, MROWS / 128), blk, 0, stream>>>(
      yhd, wpr, bpr, y);
}